// ESGATLayer_8735963480442
// MI455X (gfx1250) — compile-verified
//
#include <hip/hip_runtime.h>

// Problem constants (match reference)
#define N_SRC   100000
#define N_DST   100000
#define N_EDGE  1000000
#define IN_DIM  128
#define OUT_DIM 64
#define FEAT_DIM 64

typedef __attribute__((ext_vector_type(2))) float v2f;
typedef __attribute__((ext_vector_type(8))) float v8f;

// ---------------------------------------------------------------------------
// Sign-aware float atomic max (works with init value -inf).
// ---------------------------------------------------------------------------
__device__ __forceinline__ void atomicMaxFloat(float* addr, float val) {
    if (val >= 0.0f) {
        atomicMax((int*)addr, __float_as_int(val));
    } else {
        atomicMin((unsigned int*)addr, __float_as_uint(val));
    }
}

// ---------------------------------------------------------------------------
// Init: zero output h, m = -inf, s = 0, and wf_wa[j] = sum_k Wf[k][j]*Wa[2D+k]
// (fuses the tfidf GEMM: a_e = tfidf @ (Wf @ Wa[2D:]))
// ---------------------------------------------------------------------------
__global__ void init_kernel(float* __restrict__ h, float* __restrict__ m,
                            float* __restrict__ s, float* __restrict__ wf_wa,
                            const float* __restrict__ Wf,
                            const float* __restrict__ Wa) {
    long idx = (long)blockIdx.x * blockDim.x + threadIdx.x;
    const long total = (long)N_DST * OUT_DIM;
    if (idx < total) h[idx] = 0.0f;
    if (idx < N_DST) {
        m[idx] = -__builtin_huge_valf();
        s[idx] = 0.0f;
    }
    if (idx < OUT_DIM) {
        float acc = 0.0f;
        #pragma unroll 8
        for (int k = 0; k < FEAT_DIM; ++k)
            acc += Wf[k * OUT_DIM + idx] * Wa[2 * OUT_DIM + k];
        wf_wa[idx] = acc;
    }
}

// ---------------------------------------------------------------------------
// Z = X @ W, X:[nrows, IN_DIM] f32, W:[IN_DIM, OUT_DIM] f32, via
// V_WMMA_F32_16X16X4_F32.  One wave per 16x16 output tile; 4 waves/block
// cover the 4 column tiles (OUT_DIM=64).  K-loop: 32 WMMAs of K=4.
//
// VGPR layouts (ISA 7.12.2, wave32):
//   A 16x4 f32 : lane = M%16 (+16 -> K-half); vgpr0=K0/K2, vgpr1=K1/K3
//   B 4x16 f32 : lane = N%16 (+16 -> K-half); vgpr0=K0/K2, vgpr1=K1/K3
//   C/D 16x16  : 8 VGPRs; vgpr i: lanes 0-15 -> M=i, lanes 16-31 -> M=8+i
// ---------------------------------------------------------------------------
__global__ __launch_bounds__(128) void gemm_wmma_f32(
        const float* __restrict__ X, const float* __restrict__ W,
        float* __restrict__ Z, int nrows) {
    const int lane = threadIdx.x & 31;
    const int wave = threadIdx.x >> 5;       // 0..3 -> column tile
    const int m0 = blockIdx.x * 16;
    const int n0 = wave * 16;
    if (m0 >= nrows) return;                 // wave-uniform guard

    const int kh  = lane >> 4;               // 0: K0/K1, 1: K2/K3
    const int l15 = lane & 15;

    const float* arow = X + (long)(m0 + l15) * IN_DIM + 2 * kh;
    const float* bcol = W + (long)(2 * kh) * OUT_DIM + (n0 + l15);

    v8f c = {};
    #pragma unroll 4
    for (int k = 0; k < IN_DIM; k += 4) {
        v2f a, b;
        a[0] = arow[k];                      // X[m0+l15][k + 2*kh]
        a[1] = arow[k + 1];                  // X[m0+l15][k + 2*kh + 1]
        b[0] = bcol[(long)k * OUT_DIM];      // W[k + 2*kh][n0+l15]
        b[1] = bcol[(long)(k + 1) * OUT_DIM];
        c = __builtin_amdgcn_wmma_f32_16x16x4_f32(
                /*neg_a=*/false, a, /*neg_b=*/false, b,
                /*c_mod=*/(short)0, c, /*reuse_a=*/false, /*reuse_b=*/false);
    }
    #pragma unroll
    for (int i = 0; i < 8; ++i) {
        int row = m0 + i + 8 * kh;
        Z[(long)row * OUT_DIM + n0 + l15] = c[i];
    }
}

// ---------------------------------------------------------------------------
// a[row] = dot(Z[row,:], wa[0:64]).  Wave per row, 2 elems/lane, shfl reduce.
// ---------------------------------------------------------------------------
__global__ __launch_bounds__(256) void rowdot_kernel(
        const float* __restrict__ Z, const float* __restrict__ wa,
        float* __restrict__ out, int nrows) {
    const int wave = threadIdx.x >> 5;
    const int lane = threadIdx.x & 31;
    const int row = blockIdx.x * 8 + wave;
    if (row >= nrows) return;
    const float* zr = Z + (long)row * OUT_DIM;
    float v = zr[lane] * wa[lane] + zr[lane + 32] * wa[lane + 32];
    #pragma unroll
    for (int off = 16; off > 0; off >>= 1)
        v += __shfl_xor(v, off, 32);
    if (lane == 0) out[row] = v;
}

// ---------------------------------------------------------------------------
// Edge pass 1: e = leaky_relu(a_src[s] + a_dst[d] + tfidf[e].wf_wa, 0.01)
//              store e; atomic max into m[d].  Wave per edge.
// ---------------------------------------------------------------------------
__global__ __launch_bounds__(256) void edge_logits_kernel(
        const float* __restrict__ tfidf, const float* __restrict__ wf_wa,
        const int* __restrict__ src_idx, const int* __restrict__ dst_idx,
        const float* __restrict__ a_src, const float* __restrict__ a_dst,
        float* __restrict__ e_buf, float* __restrict__ m) {
    const int wave = threadIdx.x >> 5;
    const int lane = threadIdx.x & 31;
    const long e = (long)blockIdx.x * 8 + wave;
    if (e >= N_EDGE) return;
    const float* tf = tfidf + e * FEAT_DIM;
    float v = tf[lane] * wf_wa[lane] + tf[lane + 32] * wf_wa[lane + 32];
    #pragma unroll
    for (int off = 16; off > 0; off >>= 1)
        v += __shfl_xor(v, off, 32);
    if (lane == 0) {
        const int si = src_idx[e];
        const int di = dst_idx[e];
        float logit = v + a_src[si] + a_dst[di];
        logit = logit > 0.0f ? logit : 0.01f * logit;   // leaky_relu
        e_buf[e] = logit;
        atomicMaxFloat(&m[di], logit);
    }
}

// ---------------------------------------------------------------------------
// Edge pass 2: s[d] += exp(e - m[d]).  Thread per edge (L2-resident atomics).
// ---------------------------------------------------------------------------
__global__ __launch_bounds__(256) void edge_expsum_kernel(
        const float* __restrict__ e_buf, const int* __restrict__ dst_idx,
        const float* __restrict__ m, float* __restrict__ s) {
    const long e = (long)blockIdx.x * blockDim.x + threadIdx.x;
    if (e >= N_EDGE) return;
    const int di = dst_idx[e];
    atomicAdd(&s[di], __expf(e_buf[e] - m[di]));
}

// ---------------------------------------------------------------------------
// Edge pass 3: alpha = exp(e - m[d]) / s[d];  h[d,:] += alpha * z_src[s,:]
// Wave per edge, 2 floats/lane, coalesced scatter-add (stays in L2).
// ---------------------------------------------------------------------------
__global__ __launch_bounds__(256) void edge_scatter_kernel(
        const float* __restrict__ e_buf, const int* __restrict__ src_idx,
        const int* __restrict__ dst_idx, const float* __restrict__ m,
        const float* __restrict__ s, const float* __restrict__ z_src,
        float* __restrict__ h) {
    const int wave = threadIdx.x >> 5;
    const int lane = threadIdx.x & 31;
    const long e = (long)blockIdx.x * 8 + wave;
    if (e >= N_EDGE) return;
    const int si = src_idx[e];
    const int di = dst_idx[e];
    const float alpha = __expf(e_buf[e] - m[di]) / s[di];
    const float* zs = z_src + (long)si * OUT_DIM;
    float* hd = h + (long)di * OUT_DIM;
    atomicAdd(&hd[lane],      alpha * zs[lane]);
    atomicAdd(&hd[lane + 32], alpha * zs[lane + 32]);
}

// ---------------------------------------------------------------------------
extern "C" void kernel_launch(void* const* d_in, const int* in_sizes, int n_in,
                              void* d_out, int out_size, void* d_ws, size_t ws_size,
                              hipStream_t stream) {
    const float* origin   = (const float*)d_in[0];
    const float* neighbor = (const float*)d_in[1];
    const float* tfidf    = (const float*)d_in[2];
    const int*   src_idx  = (const int*)  d_in[3];
    const int*   dst_idx  = (const int*)  d_in[4];
    const float* Wo       = (const float*)d_in[5];
    const float* Wn       = (const float*)d_in[6];
    const float* Wf       = (const float*)d_in[7];
    const float* Wa       = (const float*)d_in[8];
    float* h = (float*)d_out;

    // Workspace carve-up (floats): ~56.8 MB total
    float* ws    = (float*)d_ws;
    float* z_src = ws;                                  // N_SRC*OUT_DIM
    float* z_dst = z_src + (size_t)N_SRC * OUT_DIM;     // N_DST*OUT_DIM
    float* a_src = z_dst + (size_t)N_DST * OUT_DIM;     // N_SRC
    float* a_dst = a_src + N_SRC;                       // N_DST
    float* mbuf  = a_dst + N_DST;                       // N_DST
    float* sbuf  = mbuf + N_DST;                        // N_DST
    float* e_buf = sbuf + N_DST;                        // N_EDGE
    float* wf_wa = e_buf + N_EDGE;                      // OUT_DIM

    const long htotal = (long)N_DST * OUT_DIM;
    init_kernel<<<(unsigned)((htotal + 255) / 256), 256, 0, stream>>>(
        h, mbuf, sbuf, wf_wa, Wf, Wa);

    // WMMA GEMMs: 6250 row-tiles, 4 column-tile waves per block
    gemm_wmma_f32<<<N_SRC / 16, 128, 0, stream>>>(neighbor, Wn, z_src, N_SRC);
    gemm_wmma_f32<<<N_DST / 16, 128, 0, stream>>>(origin,   Wo, z_dst, N_DST);

    rowdot_kernel<<<(N_SRC + 7) / 8, 256, 0, stream>>>(z_src, Wa,           a_src, N_SRC);
    rowdot_kernel<<<(N_DST + 7) / 8, 256, 0, stream>>>(z_dst, Wa + OUT_DIM, a_dst, N_DST);

    edge_logits_kernel<<<(N_EDGE + 7) / 8, 256, 0, stream>>>(
        tfidf, wf_wa, src_idx, dst_idx, a_src, a_dst, e_buf, mbuf);
    edge_expsum_kernel<<<(N_EDGE + 255) / 256, 256, 0, stream>>>(
        e_buf, dst_idx, mbuf, sbuf);
    edge_scatter_kernel<<<(N_EDGE + 7) / 8, 256, 0, stream>>>(
        e_buf, src_idx, dst_idx, mbuf, sbuf, z_src, h);
}